// SelfExpression_15899968930166
// MI455X (gfx1250) — compile-verified
//
#include <hip/hip_runtime.h>
#include <hip/hip_bf16.h>

typedef __attribute__((ext_vector_type(2))) float v2f;
typedef __attribute__((ext_vector_type(2))) int   v2i;
typedef __attribute__((ext_vector_type(8))) float v8f;

#define NN 16384
#define DD 32
#define WAVES_PER_BLOCK 8
#define KCHUNK 32          // one buffered chunk = 8 WMMA K-steps
#define PF_DIST 2048       // prefetch distance in floats (8 KB, past next chunk)

// zs = relu(Coefficient * (knn>0)) @ z
// One wave computes a 16-row x 32-col output tile via two f32 WMMA
// accumulators (16x16 each). K-loop software-pipelined with two
// statically-indexed register buffers so a full 32-K chunk of loads stays
// in flight while the previous chunk feeds V_WMMA_F32_16X16X4_F32.
__global__ __launch_bounds__(256) void SelfExpression_wmma_kernel(
    const float* __restrict__ z,      // [NN, DD] row-major
    const float* __restrict__ coef,   // [NN, NN] row-major
    const int*   __restrict__ knn,    // [NN, NN] row-major
    float*       __restrict__ out)    // [NN, DD] row-major
{
    const int lane = threadIdx.x & 31;
    const int wave = threadIdx.x >> 5;
    const int half = lane >> 4;      // 0: lanes 0-15, 1: lanes 16-31
    const int ln   = lane & 15;

    const long rowBase = ((long)blockIdx.x * WAVES_PER_BLOCK + wave) * 16;
    const long m       = rowBase + ln;          // A-matrix row for this lane

    const float* __restrict__ crow = coef + m * (long)NN;
    const int*   __restrict__ krow = knn  + m * (long)NN;

    v8f c0 = {};  // output cols 0..15
    v8f c1 = {};  // output cols 16..31

    // Double-buffered chunk operands (static indices only -> stay in VGPRs).
    v2f cb[2][8];                // coef pairs
    v2i kb[2][8];                // knn pairs
    v2f zb0[2][8], zb1[2][8];    // z fragments, cols 0-15 / 16-31

    // ---- chunk loader: issues one long load clause for chunk at k0 ----
    auto load_chunk = [&](int buf, int k0) {
        __builtin_prefetch(crow + k0 + PF_DIST, 0, 0);
        __builtin_prefetch(krow + k0 + PF_DIST, 0, 0);
        #pragma unroll
        for (int s = 0; s < 8; ++s) {
            const int koff = k0 + s * 4 + half * 2;  // f32 A layout: half h -> K=2h,2h+1
            cb[buf][s] = *reinterpret_cast<const v2f*>(crow + koff);
            kb[buf][s] = *reinterpret_cast<const v2i*>(krow + koff);
            const float* __restrict__ zr = z + (long)koff * DD;
            zb0[buf][s].x = zr[ln];            // z[koff][ln]
            zb0[buf][s].y = zr[DD + ln];       // z[koff+1][ln]
            zb1[buf][s].x = zr[ln + 16];       // z[koff][ln+16]
            zb1[buf][s].y = zr[DD + ln + 16];  // z[koff+1][ln+16]
        }
    };

    // ---- chunk consumer: relu+mask (branch-free) then 2 WMMAs per step ----
    auto compute_chunk = [&](int buf) {
        #pragma unroll
        for (int s = 0; s < 8; ++s) {
            const float r0 = fmaxf(cb[buf][s].x, 0.0f);
            const float r1 = fmaxf(cb[buf][s].y, 0.0f);
            const int q0 = -(int)(kb[buf][s].x > 0);  // all-ones or zero
            const int q1 = -(int)(kb[buf][s].y > 0);
            v2f a;
            a.x = __int_as_float(__float_as_int(r0) & q0);
            a.y = __int_as_float(__float_as_int(r1) & q1);
            c0 = __builtin_amdgcn_wmma_f32_16x16x4_f32(
                    false, a, false, zb0[buf][s], (short)0, c0, false, false);
            c1 = __builtin_amdgcn_wmma_f32_16x16x4_f32(
                    false, a, false, zb1[buf][s], (short)0, c1, false, false);
        }
    };

    // ---- software pipeline: buffers alternate with static indices ----
    load_chunk(0, 0);
    for (int k0 = 0; k0 < NN; k0 += 2 * KCHUNK) {
        load_chunk(1, k0 + KCHUNK);           // overlap with compute of buf 0
        compute_chunk(0);
        if (k0 + 2 * KCHUNK < NN)             // uniform scalar guard, no OOB
            load_chunk(0, k0 + 2 * KCHUNK);   // overlap with compute of buf 1
        compute_chunk(1);
    }

    // --- store: lane l writes rows (v + 8*half), col ln and ln+16 ---
    float* __restrict__ orow = out + (rowBase + half * 8) * DD;
    #pragma unroll
    for (int v = 0; v < 8; ++v) {
        orow[v * DD + ln]      = c0[v];
        orow[v * DD + 16 + ln] = c1[v];
    }
}

extern "C" void kernel_launch(void* const* d_in, const int* in_sizes, int n_in,
                              void* d_out, int out_size, void* d_ws, size_t ws_size,
                              hipStream_t stream) {
    const float* z    = (const float*)d_in[0];  // [NN, DD] fp32
    const float* coef = (const float*)d_in[1];  // [NN, NN] fp32
    const int*   knn  = (const int*)  d_in[2];  // [NN, NN] int32
    float*       out  = (float*)d_out;          // [NN, DD] fp32

    dim3 grid(NN / (16 * WAVES_PER_BLOCK));  // 128 blocks
    dim3 block(32 * WAVES_PER_BLOCK);        // 256 threads = 8 waves
    SelfExpression_wmma_kernel<<<grid, block, 0, stream>>>(z, coef, knn, out);
}